// GWDLoss_81346680586748
// MI455X (gfx1250) — compile-verified
//
#include <hip/hip_runtime.h>
#include <math.h>
#include <stdint.h>

// ---------------------------------------------------------------------------
// GWD loss for ellipse detection (argmax-gather-loss), MI455X / gfx1250.
//
// Bulk work = per-batch argmax over 16384 floats (64 MiB total) -> HBM-bound,
// ~2.9us at 23.3 TB/s. No matmul structure -> WMMA not applicable; instead we
// use the CDNA5 async global->LDS path (GLOBAL_LOAD_ASYNC_TO_LDS_B128 +
// s_wait_asynccnt) to keep a deep per-wave pipeline of b128 loads in flight
// with no VGPR residency for in-flight data.
// ---------------------------------------------------------------------------

#define ASYNC_B128_TO_LDS(ldsaddr, goff, sbase)                         \
  asm volatile("global_load_async_to_lds_b128 %0, %1, %2"               \
               :: "v"(ldsaddr), "v"(goff), "s"(sbase) : "memory")

__global__ __launch_bounds__(256) void gwd_stage1(
    const float* __restrict__ hm,      // (B, 1, H, W)
    const float* __restrict__ ab,      // (B, 2, H, W)
    const float* __restrict__ trig,    // (B, 2, H, W)  ch0=sin2A ch1=cos2A
    const float* __restrict__ center,  // (B, 2)
    const float* __restrict__ target,  // (B, 5) xywhr (deg)
    float* __restrict__ ws,            // (B,) per-batch loss
    int HW, int nStages)               // nStages = HW / 1024
{
  // 4-slot ring: 4 x (256 lanes x 16B) = 16 KiB
  __shared__ float4 lbuf[1024];
  __shared__ float  s_val[8];
  __shared__ int    s_idx[8];

  const int b = blockIdx.x;
  const int t = threadIdx.x;
  const float* hm_b = hm + (size_t)b * (size_t)HW;
  // addrspace(3) offset: low 32 bits of the flat shared pointer
  const unsigned lds_base = (unsigned)(uintptr_t)lbuf;

  auto issue = [&](int st) {
    unsigned goff = (unsigned)(st * 4096 + t * 16);                    // bytes
    unsigned la   = lds_base + (unsigned)(((st & 3) * 256 + t) * 16);  // bytes
    ASYNC_B128_TO_LDS(la, goff, hm_b);
  };

  // prologue: 3 stages in flight per wave (ASYNCcnt tracks per-wave issues)
  for (int p = 0; p < 3 && p < nStages; ++p) issue(p);

  float bestv = -INFINITY;
  int   besti = 0;

#pragma unroll 1
  for (int s = 0; s < nStages; ++s) {
    // oldest outstanding stage is s; outstanding count = min(3, nStages - s)
    int rem = nStages - 1 - s;
    if (rem >= 2)      asm volatile("s_wait_asynccnt 2" ::: "memory");
    else if (rem == 1) asm volatile("s_wait_asynccnt 1" ::: "memory");
    else               asm volatile("s_wait_asynccnt 0" ::: "memory");

    float4 v = lbuf[(s & 3) * 256 + t];   // ds_load_b128, own wave's slot only
    int base = s * 1024 + t * 4;
    if (v.x > bestv) { bestv = v.x; besti = base;     }
    if (v.y > bestv) { bestv = v.y; besti = base + 1; }
    if (v.z > bestv) { bestv = v.z; besti = base + 2; }
    if (v.w > bestv) { bestv = v.w; besti = base + 3; }

    // refill the slot consumed last iteration ((s+3)&3 == (s-1)&3, not the slot just read)
    if (s + 3 < nStages) issue(s + 3);
  }

  // wave32 argmax reduce, first-index wins on ties (matches jnp.argmax)
  for (int off = 16; off; off >>= 1) {
    float ov = __shfl_down(bestv, off, 32);
    int   oi = __shfl_down(besti, off, 32);
    if (ov > bestv || (ov == bestv && oi < besti)) { bestv = ov; besti = oi; }
  }
  const int wave = t >> 5, lane = t & 31;
  if (lane == 0) { s_val[wave] = bestv; s_idx[wave] = besti; }
  __syncthreads();

  if (t == 0) {
    for (int w = 1; w < 8; ++w) {
      if (s_val[w] > bestv || (s_val[w] == bestv && s_idx[w] < besti)) {
        bestv = s_val[w]; besti = s_idx[w];
      }
    }

    // gather 4 scalars at argmax
    const size_t bo2 = (size_t)b * 2u * (size_t)HW;
    float ab0   = ab[bo2 + besti];
    float ab1   = ab[bo2 + HW + besti];
    float sin2A = trig[bo2 + besti];
    float cos2A = trig[bo2 + HW + besti];
    float cxp = center[2 * b], cyp = center[2 * b + 1];
    const float* tg = target + (size_t)b * 5u;
    float xt = tg[0], yt = tg[1], wt = tg[2], ht = tg[3], rt_deg = tg[4];

    const float EPS = 1e-7f, MX = 1e7f;
    const float R2D = 57.29577951308232f, D2R = 0.017453292519943295f;

    float wp  = fminf(fmaxf(ab0 * 2.f, EPS), MX);
    float hp  = fminf(fmaxf(ab1 * 2.f, EPS), MX);
    float wtc = fminf(fmaxf(wt, EPS), MX);
    float htc = fminf(fmaxf(ht, EPS), MX);

    float rp = atan2f(sin2A, cos2A) * 0.5f;
    rp = (rp * R2D) * D2R;          // faithful rad2deg->deg2rad roundtrip
    float rt = rt_deg * D2R;

    float ap = 0.5f * wp, bp = 0.5f * hp, at = 0.5f * wtc, bt = 0.5f * htc;
    float dx = cxp - xt, dy = cyp - yt;
    float xyd = dx * dx + dy * dy;

    // trace(Sigma_p * Sigma_t) depends only on delta = rt - rp:
    float dlt = rt - rp;
    float cd = cosf(dlt), sd = sinf(dlt);
    float c2 = cd * cd, s2 = sd * sd;
    float ap2 = ap * ap, bp2 = bp * bp, at2 = at * at, bt2 = bt * bt;
    float tr   = ap2 * (c2 * at2 + s2 * bt2) + bp2 * (s2 * at2 + c2 * bt2);
    float dets = sqrtf(fmaxf(ap * bp * at * bt, 0.f));
    float whr  = (ap2 + bp2) + (at2 + bt2)
               - 2.f * sqrtf(fmaxf(tr + 2.f * dets, 0.f));
    float dist = fmaxf(xyd + whr, 0.f);
    ws[b] = 1.f - 1.f / (1.f + dist);
  }
}

__global__ __launch_bounds__(256) void gwd_stage2(
    const float* __restrict__ ws, float* __restrict__ out, int n)
{
  __shared__ float sh[8];
  float acc = 0.f;
  for (int i = threadIdx.x; i < n; i += 256) acc += ws[i];
  for (int off = 16; off; off >>= 1) acc += __shfl_down(acc, off, 32);
  if ((threadIdx.x & 31) == 0) sh[threadIdx.x >> 5] = acc;
  __syncthreads();
  if (threadIdx.x == 0) {
    float s = 0.f;
    for (int w = 0; w < 8; ++w) s += sh[w];
    out[0] = s / (float)n;
  }
}

extern "C" void kernel_launch(void* const* d_in, const int* in_sizes, int n_in,
                              void* d_out, int out_size, void* d_ws, size_t ws_size,
                              hipStream_t stream) {
  const float* hm     = (const float*)d_in[0];  // (B,1,H,W)
  const float* ab     = (const float*)d_in[1];  // (B,2,H,W)
  const float* trig   = (const float*)d_in[2];  // (B,2,H,W)
  const float* center = (const float*)d_in[3];  // (B,2)
  const float* target = (const float*)d_in[4];  // (B,5)
  float* out = (float*)d_out;
  float* ws  = (float*)d_ws;   // B floats of scratch

  const int B  = in_sizes[3] / 2;
  const int HW = in_sizes[0] / B;          // 16384; assumed multiple of 1024
  const int nStages = HW / 1024;

  gwd_stage1<<<B, 256, 0, stream>>>(hm, ab, trig, center, target, ws, HW, nStages);
  gwd_stage2<<<1, 256, 0, stream>>>(ws, out, B);
}